// SphericalHMC_79594333929598
// MI455X (gfx1250) — compile-verified
//
#include <hip/hip_runtime.h>

#define BTOT   131072
#define DIM    32
#define NSTEP  16
#define EPS    0.05f
#define BLOCK  256
#define TSTRIDE 34            // theta row stride in LDS dwords (conflict-free, 8B chunk aligned)
#define INV2PI 0.15915494309189535f

typedef float v4f __attribute__((ext_vector_type(4)));   // native vector for nontemporal store

// 4-way split dot products: shortens serial FMA chains 33 -> ~9 for better ILP.
__device__ __forceinline__ float dot_self32(const float (&x)[DIM + 1]) {
    float a0 = 0.f, a1 = 0.f, a2 = 0.f, a3 = 0.f;
#pragma unroll
    for (int i = 0; i < DIM; i += 4) {
        a0 = __builtin_fmaf(x[i],     x[i],     a0);
        a1 = __builtin_fmaf(x[i + 1], x[i + 1], a1);
        a2 = __builtin_fmaf(x[i + 2], x[i + 2], a2);
        a3 = __builtin_fmaf(x[i + 3], x[i + 3], a3);
    }
    return (a0 + a1) + (a2 + a3);
}

__device__ __forceinline__ float dot_self33(const float (&x)[DIM + 1]) {
    float a0 = x[DIM] * x[DIM], a1 = 0.f, a2 = 0.f, a3 = 0.f;
#pragma unroll
    for (int i = 0; i < DIM; i += 4) {
        a0 = __builtin_fmaf(x[i],     x[i],     a0);
        a1 = __builtin_fmaf(x[i + 1], x[i + 1], a1);
        a2 = __builtin_fmaf(x[i + 2], x[i + 2], a2);
        a3 = __builtin_fmaf(x[i + 3], x[i + 3], a3);
    }
    return (a0 + a1) + (a2 + a3);
}

__device__ __forceinline__ void kick(float (&ta)[DIM + 1], float (&v)[DIM + 1], float k) {
    float t2 = dot_self32(ta);
    float a = k * (1.0f - t2);
#pragma unroll
    for (int i = 0; i < DIM; ++i) v[i] = __builtin_fmaf(-ta[i], a, v[i]);
    v[DIM] = __builtin_fmaf(ta[DIM], k * t2, v[DIM]);
}

__device__ __forceinline__ void rotate_step(float (&ta)[DIM + 1], float (&v)[DIM + 1]) {
    float v2  = dot_self33(v);
    float rin = __builtin_amdgcn_rsqf(v2);          // v_rsq_f32
    float vn  = v2 * rin;                           // ||v||
    float ang = vn * (EPS * INV2PI);                // normalized for v_sin/v_cos
    float c   = __builtin_amdgcn_cosf(ang);         // v_cos_f32
    float s   = __builtin_amdgcn_sinf(ang);         // v_sin_f32
    float svn = s * rin;                            // s / ||v||
    float nvs = vn * s;                             // ||v|| * s
#pragma unroll
    for (int i = 0; i <= DIM; ++i) {
        float tn = __builtin_fmaf(v[i], svn, ta[i] * c);
        v[i]     = __builtin_fmaf(-ta[i], nvs, v[i] * c);
        ta[i]    = tn;
    }
}

__global__ __launch_bounds__(BLOCK) void spherical_hmc_kernel(
        const float* __restrict__ theta0,
        const float* __restrict__ vnoise,
        const float* __restrict__ accept_u,
        float* __restrict__ out) {
    __shared__ __attribute__((aligned(16))) float sT[BLOCK * TSTRIDE];   // 34 KB
    __shared__ __attribute__((aligned(16))) float sV[BLOCK * (DIM + 1)]; // 33 KB

    const unsigned tid = threadIdx.x;
    const unsigned bid = blockIdx.x;
    const float* gT = theta0 + (size_t)bid * BLOCK * DIM;
    const float* gV = vnoise + (size_t)bid * BLOCK * (DIM + 1);

    const unsigned sTbase = (unsigned)(size_t)(&sT[0]);   // low 32 bits of LDS flat addr
    const unsigned sVbase = (unsigned)(size_t)(&sV[0]);

    // ---- Async global -> LDS staging (CDNA5 ASYNCcnt path, fully coalesced) ----
    // theta tile: 256x32 dwords = 4096 b64 chunks (16/thread), repacked to stride-34 rows.
    // (LDS delta/iter 2176B != global delta 2048B, so addresses are computed.)
#pragma unroll
    for (int k = 0; k < 16; ++k) {
        unsigned c   = tid + (unsigned)k * BLOCK;
        unsigned row = c >> 4;                 // 16 b64 chunks per 32-dword row
        unsigned col = (c & 15u) << 1;         // dword within row
        unsigned lds = sTbase + ((row * TSTRIDE + col) << 2);
        unsigned long long ga = (unsigned long long)(size_t)(gT + 2u * c);
        asm volatile("global_load_async_to_lds_b64 %0, %1, off"
                     :: "v"(lds), "v"(ga) : "memory");
    }
    // v_noise tile: contiguous 33792B copy as 2112 b128 chunks. INST_OFFSET is added to
    // BOTH the LDS and global address, and the per-iteration delta (4096B) is identical,
    // so one VGPR address pair + immediate offsets covers all 8 rounds.
    {
        unsigned lds0 = sVbase + (tid << 4);
        unsigned long long ga0 = (unsigned long long)(size_t)(gV + 4u * tid);
        asm volatile("global_load_async_to_lds_b128 %0, %1, off offset:0\n\t"
                     "global_load_async_to_lds_b128 %0, %1, off offset:4096\n\t"
                     "global_load_async_to_lds_b128 %0, %1, off offset:8192\n\t"
                     "global_load_async_to_lds_b128 %0, %1, off offset:12288\n\t"
                     "global_load_async_to_lds_b128 %0, %1, off offset:16384\n\t"
                     "global_load_async_to_lds_b128 %0, %1, off offset:20480\n\t"
                     "global_load_async_to_lds_b128 %0, %1, off offset:24576\n\t"
                     "global_load_async_to_lds_b128 %0, %1, off offset:28672"
                     :: "v"(lds0), "v"(ga0) : "memory");
        if (tid < 64u) {   // remainder 64 chunks (2112 - 8*256)
            asm volatile("global_load_async_to_lds_b128 %0, %1, off offset:32768"
                         :: "v"(lds0), "v"(ga0) : "memory");
        }
    }
    asm volatile("s_wait_asynccnt 0x0" ::: "memory");
    __syncthreads();

    // ---- Register-resident chain state ----
    const float* rowT = &sT[tid * TSTRIDE];
    const float* rowV = &sV[tid * (DIM + 1)];

    float ta[DIM + 1], v[DIM + 1];
#pragma unroll
    for (int i = 0; i < DIM; ++i) ta[i] = rowT[i];
    float t2 = dot_self32(ta);
    ta[DIM] = __builtin_amdgcn_sqrtf(1.0f - t2);

    float d0 = 0.f, d1 = 0.f, d2 = 0.f, d3 = 0.f;
#pragma unroll
    for (int i = 0; i < DIM; i += 4) {
        v[i]     = rowV[i];     d0 = __builtin_fmaf(ta[i],     v[i],     d0);
        v[i + 1] = rowV[i + 1]; d1 = __builtin_fmaf(ta[i + 1], v[i + 1], d1);
        v[i + 2] = rowV[i + 2]; d2 = __builtin_fmaf(ta[i + 2], v[i + 2], d2);
        v[i + 3] = rowV[i + 3]; d3 = __builtin_fmaf(ta[i + 3], v[i + 3], d3);
    }
    v[DIM] = rowV[DIM];
    float d = ((d0 + d1) + (d2 + d3)) + ta[DIM] * v[DIM];
#pragma unroll
    for (int i = 0; i <= DIM; ++i) v[i] = __builtin_fmaf(-ta[i], d, v[i]);
    float h0 = 0.5f * (t2 + dot_self33(v));

    // fused leapfrog: half-kick, 15x(rotate + full kick), rotate + final half-kick
    kick(ta, v, 0.5f * EPS);
    for (int l = 0; l < NSTEP - 1; ++l) {
        rotate_step(ta, v);
        kick(ta, v, EPS);
    }
    rotate_step(ta, v);

    float t2f = dot_self32(ta);
    float a = 0.5f * EPS * (1.0f - t2f);
#pragma unroll
    for (int i = 0; i < DIM; ++i) v[i] = __builtin_fmaf(-ta[i], a, v[i]);
    v[DIM] = __builtin_fmaf(ta[DIM], 0.5f * EPS * t2f, v[DIM]);

    float h = 0.5f * (t2f + dot_self33(v));

    const unsigned gid = bid * BLOCK + tid;
    float pacc = __builtin_amdgcn_exp2f((h0 - h) * 1.4426950408889634f); // exp(h0-h)
    bool acc = accept_u[gid] < pacc;

    float* o = out + (size_t)gid * DIM;   // 128B-aligned row, write-once -> non-temporal
#pragma unroll
    for (int i = 0; i < DIM; i += 4) {
        v4f w;
        w.x = acc ? ta[i]     : rowT[i];
        w.y = acc ? ta[i + 1] : rowT[i + 1];
        w.z = acc ? ta[i + 2] : rowT[i + 2];
        w.w = acc ? ta[i + 3] : rowT[i + 3];
        __builtin_nontemporal_store(w, (v4f*)(o + i));
    }
}

extern "C" void kernel_launch(void* const* d_in, const int* in_sizes, int n_in,
                              void* d_out, int out_size, void* d_ws, size_t ws_size,
                              hipStream_t stream) {
    (void)in_sizes; (void)n_in; (void)out_size; (void)d_ws; (void)ws_size;
    const float* theta0   = (const float*)d_in[0];
    const float* vnoise   = (const float*)d_in[1];
    const float* accept_u = (const float*)d_in[2];
    float* out = (float*)d_out;
    dim3 grid(BTOT / BLOCK);
    spherical_hmc_kernel<<<grid, BLOCK, 0, stream>>>(theta0, vnoise, accept_u, out);
}